// Depth2Normal_5205500362964
// MI455X (gfx1250) — compile-verified
//
#include <hip/hip_runtime.h>
#include <hip/hip_bf16.h>
#include <math.h>

// Problem constants (from reference)
#define H_DIM 256
#define W_DIM 832
#define BATCH 16
#define EPSF  1e-8f

// Tile config: 128 x 8 pixels per 256-thread block (8 wave32s)
#define TW   128
#define TH   8
#define HALO 2
#define LW   (TW + 2 * HALO)   // 132
#define LH   (TH + 2 * HALO)   // 12

// CDNA5 async global->LDS path (gfx1250). Guarded so host pass / older
// toolchains still compile; fallback is a plain cooperative LDS copy.
#if defined(__gfx1250__) && __has_builtin(__builtin_amdgcn_global_load_async_to_lds_b32)
#define USE_ASYNC_LDS 1
#else
#define USE_ASYNC_LDS 0
#endif

#if USE_ASYNC_LDS
typedef __attribute__((address_space(1))) int g_i32;   // global int
typedef __attribute__((address_space(3))) int l_i32;   // LDS int
#endif

__global__ __launch_bounds__(256) void Depth2Normal_5205500362964_kernel(
    const float* __restrict__ depth,   // (B,1,H,W)
    const float* __restrict__ invK,    // (B,4,4)
    float* __restrict__ out)           // (B,3,H,W)
{
    __shared__ float sd[LH][LW];

    const int tid = threadIdx.x;
    const int b  = blockIdx.z;
    const int x0 = blockIdx.x * TW;
    const int y0 = blockIdx.y * TH;

    // --- 1) zero-fill LDS: exact jnp.pad zero-halo semantics at borders ---
    float* sflat = &sd[0][0];
    for (int i = tid; i < LH * LW; i += 256) sflat[i] = 0.0f;
    __syncthreads();  // ds stores visible before async engine writes LDS

    // --- 2) stage depth tile (+halo) into LDS via async global->LDS DMA ---
    const float* gdep = depth + (size_t)b * (H_DIM * W_DIM);
    for (int i = tid; i < LH * LW; i += 256) {
        const int ly = i / LW, lx = i % LW;
        const int gy = y0 + ly - HALO;
        const int gx = x0 + lx - HALO;
        if (gy >= 0 && gy < H_DIM && gx >= 0 && gx < W_DIM) {
#if USE_ASYNC_LDS
            __builtin_amdgcn_global_load_async_to_lds_b32(
                (g_i32*)(gdep + (size_t)gy * W_DIM + gx),
                (l_i32*)(sflat + i),
                /*imm offset*/ 0, /*cpol*/ 0);
#else
            sflat[i] = gdep[(size_t)gy * W_DIM + gx];
#endif
        }
    }
#if USE_ASYNC_LDS
#if __has_builtin(__builtin_amdgcn_s_wait_asynccnt)
    __builtin_amdgcn_s_wait_asynccnt(0);
#else
    asm volatile("s_wait_asynccnt 0" ::: "memory");
#endif
#endif
    __syncthreads();

    // --- 3) per-batch inv_K columns (uniform -> SGPRs) ---
    // ray(h,w) = c2 + w*c0 + h*c1, with cK = inv_K[:3, K]
    const float* K = invK + b * 16;
    const float c0x = K[0], c0y = K[4], c0z = K[8];
    const float c1x = K[1], c1y = K[5], c1z = K[9];
    const float c2x = K[2], c2y = K[6], c2z = K[10];

    const int tx = tid & (TW - 1);      // 0..127
    const int w  = x0 + tx;
    if (w >= W_DIM) return;
    const float fw = (float)w;

    const size_t plane = (size_t)H_DIM * W_DIM;
    const size_t obase_b = (size_t)b * 3 * plane;

    for (int r = (tid >> 7); r < TH; r += 2) {   // 4 pixels per thread
        const int h  = y0 + r;
        const int ly = r + HALO, lx = tx + HALO;
        const float fh = (float)h;

        // depth stencil from LDS (zeros in halo == zero xyz padding)
        const float d0   = sd[ly][lx];
        const float dxm1 = sd[ly][lx - 1];
        const float dxm2 = sd[ly][lx - 2];
        const float dxp2 = sd[ly][lx + 2];
        const float dym1 = sd[ly - 1][lx];
        const float dym2 = sd[ly - 2][lx];
        const float dyp2 = sd[ly + 2][lx];

        // center ray and point
        const float rx = c2x + fw * c0x + fh * c1x;
        const float ry = c2y + fw * c0y + fh * c1y;
        const float rz = c2z + fw * c0z + fh * c1z;
        const float px = d0 * rx, py = d0 * ry, pz = d0 * rz;

        // Reference stencil (note asymmetric inner taps: PATCH-1 = +2):
        // a  = xyz[h,w-1] - xyz[h,w+2]
        // bb = xyz[h-1,w] - xyz[h+2,w]
        // hz = xyz[h,w-2] - xyz[h,w+2]
        // vt = xyz[h-2,w] - xyz[h+2,w]
        const float ax = dxm1 * (rx - c0x) - dxp2 * (rx + 2.f * c0x);
        const float ay = dxm1 * (ry - c0y) - dxp2 * (ry + 2.f * c0y);
        const float az = dxm1 * (rz - c0z) - dxp2 * (rz + 2.f * c0z);

        const float bx = dym1 * (rx - c1x) - dyp2 * (rx + 2.f * c1x);
        const float by = dym1 * (ry - c1y) - dyp2 * (ry + 2.f * c1y);
        const float bz = dym1 * (rz - c1z) - dyp2 * (rz + 2.f * c1z);

        const float hx = dxm2 * (rx - 2.f * c0x) - dxp2 * (rx + 2.f * c0x);
        const float hy = dxm2 * (ry - 2.f * c0y) - dxp2 * (ry + 2.f * c0y);
        const float hz = dxm2 * (rz - 2.f * c0z) - dxp2 * (rz + 2.f * c0z);

        const float vx = dym2 * (rx - 2.f * c1x) - dyp2 * (rx + 2.f * c1x);
        const float vy = dym2 * (ry - 2.f * c1y) - dyp2 * (ry + 2.f * c1y);
        const float vz = dym2 * (rz - 2.f * c1z) - dyp2 * (rz + 2.f * c1z);

        // n1 = cross(a, b); n2 = cross(hz_vec, vt_vec)
        float n1x = ay * bz - az * by;
        float n1y = az * bx - ax * bz;
        float n1z = ax * by - ay * bx;
        float n2x = hy * vz - hz * vy;
        float n2y = hz * vx - hx * vz;
        float n2z = hx * vy - hy * vx;

        // orient toward viewer (flip if dot(n, xyz) > 0)
        if (n1x * px + n1y * py + n1z * pz > 0.f) { n1x = -n1x; n1y = -n1y; n1z = -n1z; }
        if (n2x * px + n2y * py + n2z * pz > 0.f) { n2x = -n2x; n2y = -n2y; n2z = -n2z; }

        // aver = normalize(normalize(n1) + normalize(n2)), re-oriented
        const float i1 = 1.0f / (sqrtf(n1x * n1x + n1y * n1y + n1z * n1z) + EPSF);
        const float i2 = 1.0f / (sqrtf(n2x * n2x + n2y * n2y + n2z * n2z) + EPSF);
        float sx = n1x * i1 + n2x * i2;
        float sy = n1y * i1 + n2y * i2;
        float sz = n1z * i1 + n2z * i2;
        const float is = 1.0f / (sqrtf(sx * sx + sy * sy + sz * sz) + EPSF);
        float avx = sx * is, avy = sy * is, avz = sz * is;
        if (avx * px + avy * py + avz * pz > 0.f) { avx = -avx; avy = -avy; avz = -avz; }

        // out is (B,3,H,W)
        const size_t o = obase_b + (size_t)h * W_DIM + w;
        out[o]             = avx;
        out[o + plane]     = avy;
        out[o + 2 * plane] = avz;
    }
}

extern "C" void kernel_launch(void* const* d_in, const int* in_sizes, int n_in,
                              void* d_out, int out_size, void* d_ws, size_t ws_size,
                              hipStream_t stream) {
    const float* depth = (const float*)d_in[0];  // (16,1,256,832) f32
    const float* invK  = (const float*)d_in[1];  // (16,4,4) f32
    float* out = (float*)d_out;                  // (16,3,256,832) f32

    dim3 block(256);
    dim3 grid((W_DIM + TW - 1) / TW,  // 7
              H_DIM / TH,             // 32
              BATCH);                 // 16
    Depth2Normal_5205500362964_kernel<<<grid, block, 0, stream>>>(depth, invK, out);
}